// GAT_28046136443436
// MI455X (gfx1250) — compile-verified
//
#include <hip/hip_runtime.h>

typedef __attribute__((ext_vector_type(2))) float v2f;
typedef __attribute__((ext_vector_type(8))) float v8f;

#define NEG_SLOPE 0.2f

// ---------------------------------------------------------------- utilities
__global__ void fillf_kernel(float* __restrict__ p, int n, float v) {
  int i = blockIdx.x * blockDim.x + threadIdx.x;
  if (i < n) p[i] = v;
}

__device__ inline void atomicMaxF(float* addr, float v) {
  // monotonic-bit-pattern trick; init value must be -inf
  if (v >= 0.0f) atomicMax((int*)addr, __float_as_int(v));
  else           atomicMin((unsigned int*)addr, __float_as_uint(v));
}

// Pack W[K][Ncol] into K-pair-major: Wp[k/2][col] = {W[k][col], W[k+1][col]}
// so the WMMA B operand (a K-pair per lane) is a single b64 load.
__global__ void pack_w_kernel(const float* __restrict__ W,
                              v2f* __restrict__ Wp, int K, int Ncol) {
  const int i = blockIdx.x * blockDim.x + threadIdx.x;   // over (K/2)*Ncol
  const int total = (K >> 1) * Ncol;
  if (i >= total) return;
  const int kp  = i / Ncol;
  const int col = i - kp * Ncol;
  v2f p;
  p[0] = W[(size_t)(2 * kp)     * Ncol + col];
  p[1] = W[(size_t)(2 * kp + 1) * Ncol + col];
  Wp[i] = p;
}

// ------------------------------------------------ fp32 WMMA GEMM (aligned)
// Requires M%16==0, Ncol%64==0, K%4==0. One wave computes a 16x64 strip:
// 4 accumulators, 1 A b64 + 4 B b64 loads feed 4 WMMAs per K-step.
// A: lane l -> M=l&15, K=2*(l>>4)+{0,1}; B mirrors with N=l&15.
// C/D: VGPR r, lanes 0-15 -> M=r, lanes 16-31 -> M=r+8.
__global__ void gemm_bias_wmma_n64(const float* __restrict__ X,
                                   const v2f* __restrict__ Wp,
                                   const float* __restrict__ bias,
                                   float* __restrict__ Y,
                                   int M, int K, int Ncol) {
  const int lane = threadIdx.x & 31;
  const int wave = threadIdx.x >> 5;
  const int ct = Ncol >> 6;                       // 64-wide column strips
  const int rt = M >> 4;
  const int tile = blockIdx.x * (blockDim.x >> 5) + wave;
  if (tile >= ct * rt) return;                    // wave-uniform exit

  const int rowBase = (tile / ct) << 4;
  const int colBase = (tile % ct) << 6;
  const int half = lane >> 4;                     // 0 or 1
  const int l15  = lane & 15;

  const float* __restrict__ ap = X + (size_t)(rowBase + l15) * K + half * 2;
  const v2f*   __restrict__ bp = Wp + (size_t)half * Ncol + colBase + l15;

  v8f cc[4] = {v8f{}, v8f{}, v8f{}, v8f{}};
  #pragma unroll 4
  for (int k0 = 0; k0 < K; k0 += 4) {
    const v2f a = *(const v2f*)(ap + k0);               // b64
    const v2f* __restrict__ br = bp + (size_t)(k0 >> 1) * Ncol;
    const v2f bv0 = br[0];                              // b64 each
    const v2f bv1 = br[16];
    const v2f bv2 = br[32];
    const v2f bv3 = br[48];
    cc[0] = __builtin_amdgcn_wmma_f32_16x16x4_f32(false, a, false, bv0, (short)0, cc[0], false, false);
    cc[1] = __builtin_amdgcn_wmma_f32_16x16x4_f32(false, a, false, bv1, (short)0, cc[1], false, false);
    cc[2] = __builtin_amdgcn_wmma_f32_16x16x4_f32(false, a, false, bv2, (short)0, cc[2], false, false);
    cc[3] = __builtin_amdgcn_wmma_f32_16x16x4_f32(false, a, false, bv3, (short)0, cc[3], false, false);
  }

  const int row0 = rowBase + half * 8;
  #pragma unroll
  for (int t = 0; t < 4; ++t) {
    const int col = colBase + t * 16 + l15;
    const float bb = bias ? bias[col] : 0.0f;
    float* __restrict__ yp = Y + (size_t)row0 * Ncol + col;
    #pragma unroll
    for (int r = 0; r < 8; ++r)
      yp[(size_t)r * Ncol] = cc[t][r] + bb;
  }
}

// ------------------------------------------ fp32 WMMA GEMM (ragged columns)
// Requires M%16==0, K%4==0; Ncol arbitrary (classifier, Ncol=40).
// Branchless OOB: clamp column address, multiply loaded B by 0/1 mask.
__global__ void gemm_bias_wmma_guard(const float* __restrict__ X,
                                     const v2f* __restrict__ Wp,
                                     const float* __restrict__ bias,
                                     float* __restrict__ Y,
                                     int M, int K, int Ncol) {
  const int lane = threadIdx.x & 31;
  const int wave = threadIdx.x >> 5;
  const int ct = (Ncol + 15) >> 4;
  const int rt = M >> 4;
  const int tile = blockIdx.x * (blockDim.x >> 5) + wave;
  if (tile >= ct * rt) return;

  const int rowBase = (tile / ct) << 4;
  const int colBase = (tile % ct) << 4;
  const int half = lane >> 4;
  const int l15  = lane & 15;
  const int bcol = colBase + l15;
  const bool bok = bcol < Ncol;
  const int bcolc = bok ? bcol : (Ncol - 1);
  const float bmask = bok ? 1.0f : 0.0f;

  const float* __restrict__ ap = X + (size_t)(rowBase + l15) * K + half * 2;
  const v2f*   __restrict__ bp = Wp + (size_t)half * Ncol + bcolc;

  v8f c = {};
  #pragma unroll 4
  for (int k0 = 0; k0 < K; k0 += 4) {
    const v2f a = *(const v2f*)(ap + k0);
    v2f b = bp[(size_t)(k0 >> 1) * Ncol];
    b[0] *= bmask;
    b[1] *= bmask;
    c = __builtin_amdgcn_wmma_f32_16x16x4_f32(false, a, false, b, (short)0, c, false, false);
  }

  if (bok) {
    const float bb = bias ? bias[bcol] : 0.0f;
    const int row0 = rowBase + half * 8;
    float* __restrict__ yp = Y + (size_t)row0 * Ncol + bcol;
    #pragma unroll
    for (int r = 0; r < 8; ++r)
      yp[(size_t)r * Ncol] = c[r] + bb;
  }
}

// ------------------------------------------------- edge phase (wave per edge)
// e = att . leaky_relu(xl[src] + xr[dst]);  m[dst] = max(m[dst], e)
// d4 = feature dim / 4 (feature rows are float4-tileable: 128/256)
__global__ void edge_logits_kernel(const int* __restrict__ ei, int E, int ET,
                                   const float* __restrict__ XL,
                                   const float* __restrict__ XR,
                                   const float* __restrict__ att,
                                   float* __restrict__ eo,
                                   float* __restrict__ mx, int d4) {
  const int wid  = (blockIdx.x * blockDim.x + threadIdx.x) >> 5;
  const int lane = threadIdx.x & 31;
  if (wid >= ET) return;
  int s, t;
  if (wid < E) { s = ei[wid]; t = ei[E + wid]; } else { s = t = wid - E; }
  const float4* __restrict__ xl = (const float4*)XL + (size_t)s * d4;
  const float4* __restrict__ xr = (const float4*)XR + (size_t)t * d4;
  const float4* __restrict__ at = (const float4*)att;
  float acc = 0.0f;
  for (int f = lane; f < d4; f += 32) {
    const float4 a = xl[f];                     // b128 gathers (L2-resident)
    const float4 b = xr[f];
    const float4 g = at[f];
    float h;
    h = a.x + b.x; h = h > 0.0f ? h : NEG_SLOPE * h; acc += h * g.x;
    h = a.y + b.y; h = h > 0.0f ? h : NEG_SLOPE * h; acc += h * g.y;
    h = a.z + b.z; h = h > 0.0f ? h : NEG_SLOPE * h; acc += h * g.z;
    h = a.w + b.w; h = h > 0.0f ? h : NEG_SLOPE * h; acc += h * g.w;
  }
  #pragma unroll
  for (int off = 16; off > 0; off >>= 1) acc += __shfl_xor(acc, off, 32);
  if (lane == 0) { eo[wid] = acc; atomicMaxF(&mx[t], acc); }
}

// p = exp(e - m[dst]);  s[dst] += p   (thread per edge)
__global__ void edge_softmax_kernel(const int* __restrict__ ei, int E, int ET,
                                    const float* __restrict__ mx,
                                    float* __restrict__ ep,
                                    float* __restrict__ sx) {
  const int i = blockIdx.x * blockDim.x + threadIdx.x;
  if (i >= ET) return;
  const int t = (i < E) ? ei[E + i] : (i - E);
  const float p = __expf(ep[i] - mx[t]);
  ep[i] = p;
  atomicAdd(&sx[t], p);
}

// out[dst] += (p/s[dst]) * xl[src]   (wave per edge; b128 gather, f32 atomics)
__global__ void edge_scatter_kernel(const int* __restrict__ ei, int E, int ET,
                                    const float* __restrict__ XL,
                                    const float* __restrict__ ep,
                                    const float* __restrict__ sx,
                                    float* __restrict__ out, int d4) {
  const int wid  = (blockIdx.x * blockDim.x + threadIdx.x) >> 5;
  const int lane = threadIdx.x & 31;
  if (wid >= ET) return;
  int s, t;
  if (wid < E) { s = ei[wid]; t = ei[E + wid]; } else { s = t = wid - E; }
  const float alpha = ep[wid] / sx[t];
  const float4* __restrict__ xl = (const float4*)XL + (size_t)s * d4;
  float* __restrict__ op = out + (size_t)t * (d4 * 4);
  for (int f = lane; f < d4; f += 32) {
    const float4 v = xl[f];
    float* o4 = op + f * 4;
    atomicAdd(o4 + 0, alpha * v.x);
    atomicAdd(o4 + 1, alpha * v.y);
    atomicAdd(o4 + 2, alpha * v.z);
    atomicAdd(o4 + 3, alpha * v.w);
  }
}

// out += bias (broadcast over rows, d power of two); optional relu
__global__ void finalize_kernel(float* __restrict__ out,
                                const float* __restrict__ bias,
                                int n, int dmask, int relu) {
  const int i = blockIdx.x * blockDim.x + threadIdx.x;
  if (i >= n) return;
  float v = out[i] + bias[i & dmask];
  if (relu) v = fmaxf(v, 0.0f);
  out[i] = v;
}

// ---------------------------------------------------------------- launcher
extern "C" void kernel_launch(void* const* d_in, const int* in_sizes, int n_in,
                              void* d_out, int out_size, void* d_ws, size_t ws_size,
                              hipStream_t stream) {
  const float* x   = (const float*)d_in[0];
  const int*   ei  = (const int*)  d_in[1];
  const float* w1l = (const float*)d_in[2];
  const float* b1l = (const float*)d_in[3];
  const float* w1r = (const float*)d_in[4];
  const float* b1r = (const float*)d_in[5];
  const float* a1  = (const float*)d_in[6];
  const float* o1  = (const float*)d_in[7];
  const float* w2l = (const float*)d_in[8];
  const float* b2l = (const float*)d_in[9];
  const float* w2r = (const float*)d_in[10];
  const float* b2r = (const float*)d_in[11];
  const float* a2  = (const float*)d_in[12];
  const float* o2  = (const float*)d_in[13];
  const float* w3l = (const float*)d_in[14];
  const float* b3l = (const float*)d_in[15];
  const float* w3r = (const float*)d_in[16];
  const float* b3r = (const float*)d_in[17];
  const float* a3  = (const float*)d_in[18];
  const float* o3  = (const float*)d_in[19];
  const float* wc  = (const float*)d_in[20];
  const float* bc  = (const float*)d_in[21];

  const int N  = in_sizes[0] / 128;
  const int E  = in_sizes[1] / 2;
  const int ET = E + N;

  // workspace (floats): 3 ping-pong node buffers (N x 256),
  // per-edge logits, per-node max & sum, 2 packed-W panels (<= 256x256 each)
  float* B0  = (float*)d_ws;
  float* B1  = B0 + (size_t)N * 256;
  float* B2  = B1 + (size_t)N * 256;
  float* EP  = B2 + (size_t)N * 256;
  float* Mx  = EP + (size_t)ET;
  float* Sx  = Mx + (size_t)N;
  v2f*   WPa = (v2f*)(Sx + (size_t)N);          // (K/2)*Ncol v2f elements
  v2f*   WPb = WPa + 32768;                     // 256*256/2 v2f

  auto packW = [&](const float* W, v2f* Wp, int k, int n) {
    const int total = (k >> 1) * n;
    pack_w_kernel<<<(total + 255) / 256, 256, 0, stream>>>(W, Wp, k, n);
  };
  auto gemmBig = [&](const float* X, const v2f* Wp, const float* b, float* Y,
                     int m, int k, int n) {           // m%16==0, n%64==0
    const int tiles = (m >> 4) * (n >> 6);
    gemm_bias_wmma_n64<<<(tiles + 3) / 4, 128, 0, stream>>>(X, Wp, b, Y, m, k, n);
  };
  auto gemmGuard = [&](const float* X, const v2f* Wp, const float* b, float* Y,
                       int m, int k, int n) {         // m%16==0, ragged n
    const int tiles = (m >> 4) * ((n + 15) / 16);
    gemm_bias_wmma_guard<<<(tiles + 3) / 4, 128, 0, stream>>>(X, Wp, b, Y, m, k, n);
  };

  // One GATv2 layer. XO doubles as xr buffer (dead after logits) and output.
  auto layer = [&](const float* Xin, int din, int dout,
                   const float* wl, const float* bl,
                   const float* wr, const float* br,
                   const float* att, const float* ob,
                   float* XL, float* XO, int relu) {
    packW(wl, WPa, din, dout);
    packW(wr, WPb, din, dout);
    gemmBig(Xin, WPa, bl, XL, N, din, dout);                     // xl = X@Wl+bl
    gemmBig(Xin, WPb, br, XO, N, din, dout);                     // xr = X@Wr+br
    fillf_kernel<<<(N + 255) / 256, 256, 0, stream>>>(Mx, N, -__builtin_huge_valf());
    fillf_kernel<<<(N + 255) / 256, 256, 0, stream>>>(Sx, N, 0.0f);
    edge_logits_kernel<<<(ET + 7) / 8, 256, 0, stream>>>(ei, E, ET, XL, XO, att,
                                                         EP, Mx, dout >> 2);
    fillf_kernel<<<(N * dout + 255) / 256, 256, 0, stream>>>(XO, N * dout, 0.0f);
    edge_softmax_kernel<<<(ET + 255) / 256, 256, 0, stream>>>(ei, E, ET, Mx, EP, Sx);
    edge_scatter_kernel<<<(ET + 7) / 8, 256, 0, stream>>>(ei, E, ET, XL, EP, Sx,
                                                          XO, dout >> 2);
    finalize_kernel<<<(N * dout + 255) / 256, 256, 0, stream>>>(XO, ob,
                                                                N * dout, dout - 1, relu);
  };

  layer(x,  128, 128, w1l, b1l, w1r, b1r, a1, o1, B0, B1, 1);
  layer(B1, 128, 128, w2l, b2l, w2r, b2r, a2, o2, B0, B2, 1);
  layer(B2, 128, 256, w3l, b3l, w3r, b3r, a3, o3, B0, B1, 0);
  packW(wc, WPa, 256, 40);
  gemmGuard(B1, WPa, bc, (float*)d_out, N, 256, 40);             // classifier
}